// Sparse_attention_28879360098670
// MI455X (gfx1250) — compile-verified
//
#include <hip/hip_runtime.h>
#include <hip/hip_bf16.h>
#include <stdint.h>

#define EPS 1e-07f
#define ROW_LEN 4096
#define THREADS 256
#define NWAVES (THREADS / 32)
#define VEC4_PER_THREAD 4   // 4 x float4 = 16 floats per thread, 256*16 = 4096

// Order-preserving map: fp32 -> u32 such that float compare == unsigned compare.
__device__ __forceinline__ unsigned f2ord(float f) {
    unsigned u = __float_as_uint(f);
    return (u & 0x80000000u) ? ~u : (u | 0x80000000u);
}
__device__ __forceinline__ float ord2f(unsigned r) {
    unsigned u = (r & 0x80000000u) ? (r ^ 0x80000000u) : ~r;
    return __uint_as_float(u);
}

__global__ __launch_bounds__(THREADS)
void topk_sparse_attn_kernel(const float* __restrict__ in,
                             const int* __restrict__ topk_ptr,
                             float* __restrict__ out) {
    __shared__ __align__(16) float s_row[ROW_LEN];     // 16 KB: the whole row
    __shared__ unsigned s_hist[THREADS];               // 256-bin radix histogram
    __shared__ unsigned s_wavesum[NWAVES];             // per-wave histogram totals
    __shared__ unsigned s_bcast[2];                    // [0]=prefix bits, [1]=k remaining
    __shared__ float    s_wsum[NWAVES];                // per-wave float partial sums

    const int tid  = threadIdx.x;
    const int lane = tid & 31;
    const int wv   = tid >> 5;
    const int row  = blockIdx.x;
    const int k    = *topk_ptr;

    const float* __restrict__ rowp = in  + (size_t)row * ROW_LEN;
    float*       __restrict__ orow = out + (size_t)row * ROW_LEN;

    // ---------- Phase 0: async DMA the row HBM -> LDS (gfx1250 ASYNCcnt path) ----------
    // Generic pointers to LDS carry the LDS byte offset in their low 32 bits.
    const unsigned lds_base = (unsigned)(uintptr_t)(&s_row[0]);
    #pragma unroll
    for (int j = 0; j < VEC4_PER_THREAD; ++j) {
        unsigned byte_off = (unsigned)(tid + j * THREADS) * 16u;  // coalesced b128 chunks
        unsigned lds_addr = lds_base + byte_off;
        asm volatile("global_load_async_to_lds_b128 %0, %1, %2"
                     :: "v"(lds_addr), "v"(byte_off), "s"(rowp)
                     : "memory");
    }
    asm volatile("s_wait_asynccnt 0" ::: "memory");  // drain this wave's async ops
    __syncthreads();                                 // make all waves' DMA visible

    // Static-shape branch from the reference: time_step <= top_k -> identity.
    if (k >= ROW_LEN) {
        #pragma unroll
        for (int j = 0; j < VEC4_PER_THREAD; ++j) {
            int idx = 4 * (tid + j * THREADS);
            *reinterpret_cast<float4*>(&orow[idx]) =
                *reinterpret_cast<const float4*>(&s_row[idx]);
        }
        return;
    }

    // ---------- Phase 1: exact k-th largest via 4-round 8-bit radix select ----------
    unsigned prefix = 0u;
    unsigned kRem   = (unsigned)k;

    #pragma unroll
    for (int r = 3; r >= 0; --r) {
        const unsigned maskHigh =
            (r == 3) ? 0u : (0xFFFFFFFFu << ((8 * (r + 1)) & 31));

        s_hist[tid] = 0u;
        __syncthreads();                             // barrier A

        #pragma unroll
        for (int j = 0; j < VEC4_PER_THREAD; ++j) {
            float4 v = *reinterpret_cast<const float4*>(&s_row[4 * (tid + j * THREADS)]);
            float c[4] = {v.x, v.y, v.z, v.w};
            #pragma unroll
            for (int e = 0; e < 4; ++e) {
                unsigned u = f2ord(c[e]);
                if ((u & maskHigh) == (prefix & maskHigh)) {
                    atomicAdd(&s_hist[(u >> (8 * r)) & 0xFFu], 1u);  // ds_add_u32
                }
            }
        }
        __syncthreads();                             // barrier B

        // Suffix scan over 256 bins, barrier-free inside each wave32:
        // wave w owns bins [32w, 32w+31]; intra-wave inclusive suffix scan via shuffles.
        unsigned sum = s_hist[tid];
        #pragma unroll
        for (int off = 1; off < 32; off <<= 1) {
            unsigned other = __shfl_down(sum, off, 32);
            sum += (lane + off < 32) ? other : 0u;
        }
        if (lane == 0) s_wavesum[wv] = sum;          // wave total = suffix at lane 0
        __syncthreads();                             // barrier C

        // Cross-wave suffix offset: total of all waves above this one (fixed order).
        unsigned crossOff = 0u;
        #pragma unroll
        for (int w = 0; w < NWAVES; ++w)
            crossOff += (w > wv) ? s_wavesum[w] : 0u;

        const unsigned sufHere = sum + crossOff;     // #elems with digit >= tid (in class)
        unsigned nb = __shfl_down(sufHere, 1, 32);
        const unsigned sufNext = (lane == 31) ? crossOff : nb;

        // Exactly one bin satisfies this (suffix sums are non-increasing).
        if (sufHere >= kRem && sufNext < kRem) {
            s_bcast[0] = prefix | ((unsigned)tid << (8 * r));
            s_bcast[1] = kRem - sufNext;
        }
        __syncthreads();                             // barrier D
        prefix = s_bcast[0];
        kRem   = s_bcast[1];
        // next round's barrier A protects s_hist/s_wavesum/s_bcast reuse
    }

    const float kth   = ord2f(prefix);  // exact k-th largest value of the row
    const float delta = kth + EPS;

    // ---------- Phase 2: deterministic row sum of relu(x - delta) ----------
    float lsum = 0.0f;
    #pragma unroll
    for (int j = 0; j < VEC4_PER_THREAD; ++j) {
        float4 v = *reinterpret_cast<const float4*>(&s_row[4 * (tid + j * THREADS)]);
        lsum += fmaxf(v.x - delta, 0.0f) + fmaxf(v.y - delta, 0.0f)
              + fmaxf(v.z - delta, 0.0f) + fmaxf(v.w - delta, 0.0f);
    }
    // wave32 butterfly reduce
    #pragma unroll
    for (int off = 16; off > 0; off >>= 1)
        lsum += __shfl_xor(lsum, off, 32);
    if (lane == 0) s_wsum[wv] = lsum;
    __syncthreads();
    float total = 0.0f;
    #pragma unroll
    for (int w = 0; w < NWAVES; ++w) total += s_wsum[w];  // fixed order -> deterministic
    const float inv = 1.0f / (total + EPS);

    // ---------- Phase 3: single coalesced HBM write ----------
    #pragma unroll
    for (int j = 0; j < VEC4_PER_THREAD; ++j) {
        int idx = 4 * (tid + j * THREADS);
        float4 v = *reinterpret_cast<const float4*>(&s_row[idx]);
        float4 o;
        o.x = fmaxf(v.x - delta, 0.0f) * inv;
        o.y = fmaxf(v.y - delta, 0.0f) * inv;
        o.z = fmaxf(v.z - delta, 0.0f) * inv;
        o.w = fmaxf(v.w - delta, 0.0f) * inv;
        *reinterpret_cast<float4*>(&orow[idx]) = o;
    }
}

extern "C" void kernel_launch(void* const* d_in, const int* in_sizes, int n_in,
                              void* d_out, int out_size, void* d_ws, size_t ws_size,
                              hipStream_t stream) {
    (void)n_in; (void)out_size; (void)d_ws; (void)ws_size;
    const float* attn = (const float*)d_in[0];
    const int*   topk = (const int*)d_in[1];
    float*       out  = (float*)d_out;
    const int rows = in_sizes[0] / ROW_LEN;   // 8192 for the reference shapes
    topk_sparse_attn_kernel<<<rows, THREADS, 0, stream>>>(attn, topk, out);
}